// GlobalCovPooling2D_66786741453068
// MI455X (gfx1250) — compile-verified
//
#include <hip/hip_runtime.h>
#include <math.h>

typedef __attribute__((ext_vector_type(2))) float v2f;
typedef __attribute__((ext_vector_type(8))) float v8f;

#define CDIM 256
#define NDIM 784   // 28*28
#define NBATCH 64

static __device__ __forceinline__ v8f wmma4(v2f a, v2f b, v8f c) {
    return __builtin_amdgcn_wmma_f32_16x16x4_f32(false, a, false, b,
                                                 (short)0, c, false, false);
}

// ---------------------------------------------------------------------------
// Row means: one wave per row of X[16384][784]
// ---------------------------------------------------------------------------
__global__ void mean_kernel(const float* __restrict__ X, float* __restrict__ mean) {
    int wave = threadIdx.x >> 5;
    int lane = threadIdx.x & 31;
    int row  = blockIdx.x * 8 + wave;          // 0 .. 64*256-1
    const float* p = X + (size_t)row * NDIM;
    float s = 0.f;
    for (int k = lane; k < NDIM; k += 32) s += p[k];
    for (int off = 16; off > 0; off >>= 1) s += __shfl_down(s, off, 32);
    if (lane == 0) mean[row] = s * (1.0f / (float)NDIM);
}

// ---------------------------------------------------------------------------
// Sigma = (1/n) * Xc * Xc^T  per batch, fp32 WMMA, K = 784.
// 4x2 register tiling: one wave owns a 64x32 output block (8 accumulators);
// each K-step loads 4 A-fragments + 2 B-fragments (all contiguous b64 pairs,
// since B = Xc^T means B[kk,n] = Xc[tj+n,kk]) and fires 8 WMMAs.
// ---------------------------------------------------------------------------
__global__ void sigma_kernel(const float* __restrict__ X,
                             const float* __restrict__ mean,
                             float* __restrict__ S) {
    int wave  = threadIdx.x >> 5;
    int lane  = threadIdx.x & 31;
    int tile  = (blockIdx.x & 3) * 8 + wave;   // 0..31  (4 x 8 grid of 64x32 blocks)
    int batch =  blockIdx.x >> 2;
    int ti = (tile >> 3) << 6;                 // 0,64,128,192
    int tj = (tile &  7) << 5;                 // 0..224

    const float* Xb = X    + (size_t)batch * CDIM * NDIM;
    const float* mb = mean + batch * CDIM;

    int mi = lane & 15;          // row-in-tile for A, col-in-tile for B
    int kh = (lane >> 4) << 1;   // 0 or 2
    float ma0 = mb[ti      + mi], ma1 = mb[ti + 16 + mi];
    float ma2 = mb[ti + 32 + mi], ma3 = mb[ti + 48 + mi];
    float mn0 = mb[tj      + mi], mn1 = mb[tj + 16 + mi];

    const float* ar0 = Xb + (size_t)(ti + mi) * NDIM + kh;
    const float* ar1 = ar0 + (size_t)16 * NDIM;
    const float* ar2 = ar0 + (size_t)32 * NDIM;
    const float* ar3 = ar0 + (size_t)48 * NDIM;
    const float* br0 = Xb + (size_t)(tj + mi) * NDIM + kh;
    const float* br1 = br0 + (size_t)16 * NDIM;

    v8f c00 = {}, c01 = {}, c10 = {}, c11 = {};
    v8f c20 = {}, c21 = {}, c30 = {}, c31 = {};
    for (int k = 0; k < NDIM; k += 4) {
        v2f a0 = { ar0[k] - ma0, ar0[k + 1] - ma0 };
        v2f a1 = { ar1[k] - ma1, ar1[k + 1] - ma1 };
        v2f a2 = { ar2[k] - ma2, ar2[k + 1] - ma2 };
        v2f a3 = { ar3[k] - ma3, ar3[k + 1] - ma3 };
        v2f b0 = { br0[k] - mn0, br0[k + 1] - mn0 };
        v2f b1 = { br1[k] - mn1, br1[k + 1] - mn1 };
        c00 = wmma4(a0, b0, c00);  c01 = wmma4(a0, b1, c01);
        c10 = wmma4(a1, b0, c10);  c11 = wmma4(a1, b1, c11);
        c20 = wmma4(a2, b0, c20);  c21 = wmma4(a2, b1, c21);
        c30 = wmma4(a3, b0, c30);  c31 = wmma4(a3, b1, c31);
    }

    float* Sb = S + (size_t)batch * CDIM * CDIM;
    const float inv_n = 1.0f / (float)NDIM;
    int mbase = (lane >> 4) << 3;
    int nlo   = lane & 15;
    for (int r = 0; r < 8; ++r) {
        int gm = ti + mbase + r;
        int gn0 = tj + nlo, gn1 = gn0 + 16;
        Sb[(size_t)(gm     ) * CDIM + gn0] = c00[r] * inv_n;
        Sb[(size_t)(gm     ) * CDIM + gn1] = c01[r] * inv_n;
        Sb[(size_t)(gm + 16) * CDIM + gn0] = c10[r] * inv_n;
        Sb[(size_t)(gm + 16) * CDIM + gn1] = c11[r] * inv_n;
        Sb[(size_t)(gm + 32) * CDIM + gn0] = c20[r] * inv_n;
        Sb[(size_t)(gm + 32) * CDIM + gn1] = c21[r] * inv_n;
        Sb[(size_t)(gm + 48) * CDIM + gn0] = c30[r] * inv_n;
        Sb[(size_t)(gm + 48) * CDIM + gn1] = c31[r] * inv_n;
    }
}

// ---------------------------------------------------------------------------
// trace[b] = sum_c S[b,c,c]
// ---------------------------------------------------------------------------
__global__ void trace_kernel(const float* __restrict__ S, float* __restrict__ trc) {
    __shared__ float sm[256];
    int b = blockIdx.x, t = threadIdx.x;
    sm[t] = S[(size_t)b * CDIM * CDIM + (size_t)t * CDIM + t];
    __syncthreads();
    for (int off = 128; off > 0; off >>= 1) {
        if (t < off) sm[t] += sm[t + off];
        __syncthreads();
    }
    if (t == 0) trc[b] = sm[0];
}

// ---------------------------------------------------------------------------
// Y0 = S / trace[b],  Z0 = I
// ---------------------------------------------------------------------------
__global__ void init_kernel(const float* __restrict__ S, const float* __restrict__ trc,
                            float* __restrict__ Y, float* __restrict__ Z) {
    size_t idx = (size_t)blockIdx.x * 256 + threadIdx.x;
    int b  = (int)(idx >> 16);
    int ij = (int)(idx & 65535);
    Y[idx] = S[idx] / trc[b];
    Z[idx] = ((ij >> 8) == (ij & 255)) ? 1.0f : 0.0f;
}

// ---------------------------------------------------------------------------
// Batched 256x256x256 fp32 WMMA GEMM:  D = scale*(A@B) + diag*I
//   (scale=-1, diag=3)  ->  3I - A@B
//   (scale=0.5, diag=0) ->  0.5 * A@B
// 4x2 register tiling (64x32 block per wave): 8 WMMAs per K-step of 4 with
// 4 b64 A-loads + 4 b32 B-loads -> 1.0 load-instr per WMMA.
// ---------------------------------------------------------------------------
__global__ void gemm256_kernel(const float* __restrict__ A,
                               const float* __restrict__ Bm,
                               float* __restrict__ D,
                               float scale, float diag) {
    int wave  = threadIdx.x >> 5;
    int lane  = threadIdx.x & 31;
    int tile  = (blockIdx.x & 3) * 8 + wave;   // 0..31
    int batch =  blockIdx.x >> 2;
    int ti = (tile >> 3) << 6;
    int tj = (tile &  7) << 5;

    const float* Ab = A  + (size_t)batch * CDIM * CDIM;
    const float* Bb = Bm + (size_t)batch * CDIM * CDIM;

    int mi = lane & 15;
    int kh = (lane >> 4) << 1;   // 0 or 2

    const float* ar0 = Ab + (size_t)(ti + mi) * CDIM + kh;   // A rows, contiguous pair
    const float* ar1 = ar0 + (size_t)16 * CDIM;
    const float* ar2 = ar0 + (size_t)32 * CDIM;
    const float* ar3 = ar0 + (size_t)48 * CDIM;
    const float* bc0 = Bb + (size_t)kh * CDIM + (tj + mi);   // B cols, stride-CDIM pair
    const float* bc1 = bc0 + 16;

    v8f c00 = {}, c01 = {}, c10 = {}, c11 = {};
    v8f c20 = {}, c21 = {}, c30 = {}, c31 = {};
    for (int k = 0; k < CDIM; k += 4) {
        size_t ko = (size_t)k * CDIM;
        v2f a0 = { ar0[k],  ar0[k + 1] };
        v2f a1 = { ar1[k],  ar1[k + 1] };
        v2f a2 = { ar2[k],  ar2[k + 1] };
        v2f a3 = { ar3[k],  ar3[k + 1] };
        v2f b0 = { bc0[ko], bc0[ko + CDIM] };
        v2f b1 = { bc1[ko], bc1[ko + CDIM] };
        c00 = wmma4(a0, b0, c00);  c01 = wmma4(a0, b1, c01);
        c10 = wmma4(a1, b0, c10);  c11 = wmma4(a1, b1, c11);
        c20 = wmma4(a2, b0, c20);  c21 = wmma4(a2, b1, c21);
        c30 = wmma4(a3, b0, c30);  c31 = wmma4(a3, b1, c31);
    }

    float* Db = D + (size_t)batch * CDIM * CDIM;
    int mbase = (lane >> 4) << 3;
    int nlo   = lane & 15;
    for (int r = 0; r < 8; ++r) {
        int gm = ti + mbase + r;
        int gn0 = tj + nlo, gn1 = gn0 + 16;
        float v;
        v = scale * c00[r]; if (gm      == gn0) v += diag; Db[(size_t)(gm     ) * CDIM + gn0] = v;
        v = scale * c01[r]; if (gm      == gn1) v += diag; Db[(size_t)(gm     ) * CDIM + gn1] = v;
        v = scale * c10[r]; if (gm + 16 == gn0) v += diag; Db[(size_t)(gm + 16) * CDIM + gn0] = v;
        v = scale * c11[r]; if (gm + 16 == gn1) v += diag; Db[(size_t)(gm + 16) * CDIM + gn1] = v;
        v = scale * c20[r]; if (gm + 32 == gn0) v += diag; Db[(size_t)(gm + 32) * CDIM + gn0] = v;
        v = scale * c21[r]; if (gm + 32 == gn1) v += diag; Db[(size_t)(gm + 32) * CDIM + gn1] = v;
        v = scale * c30[r]; if (gm + 48 == gn0) v += diag; Db[(size_t)(gm + 48) * CDIM + gn0] = v;
        v = scale * c31[r]; if (gm + 48 == gn1) v += diag; Db[(size_t)(gm + 48) * CDIM + gn1] = v;
    }
}

// ---------------------------------------------------------------------------
// out[b, triu(i,j)] = sqrt(trace[b]) * Y[b,i,j],  j >= i
// ---------------------------------------------------------------------------
__global__ void final_kernel(const float* __restrict__ Y,
                             const float* __restrict__ trc,
                             float* __restrict__ out) {
    int b = blockIdx.y;
    int i = blockIdx.x;
    int j = threadIdx.x;
    if (j < i) return;
    float s = sqrtf(trc[b]);
    int rowoff = i * CDIM - (i * (i - 1)) / 2;
    out[(size_t)b * 32896 + rowoff + (j - i)] =
        s * Y[(size_t)b * CDIM * CDIM + (size_t)i * CDIM + j];
}

// ---------------------------------------------------------------------------
extern "C" void kernel_launch(void* const* d_in, const int* in_sizes, int n_in,
                              void* d_out, int out_size, void* d_ws, size_t ws_size,
                              hipStream_t stream) {
    (void)in_sizes; (void)n_in; (void)out_size; (void)ws_size;

    const float* x   = (const float*)d_in[0];
    float*       out = (float*)d_out;
    float*       ws  = (float*)d_ws;

    const size_t MAT = (size_t)NBATCH * CDIM * CDIM;   // 4,194,304 floats

    float* mean = ws;                    // 64*256
    float* trc  = mean + NBATCH * CDIM;  // 64
    float* S    = trc + NBATCH;          // MAT  (reused as T after init)
    float* Ya   = S  + MAT;
    float* Yb   = Ya + MAT;
    float* Za   = Yb + MAT;
    float* Zb   = Za + MAT;

    mean_kernel <<<(NBATCH * CDIM) / 8, 256, 0, stream>>>(x, mean);
    sigma_kernel<<<NBATCH * 4,         256, 0, stream>>>(x, mean, S);
    trace_kernel<<<NBATCH,             256, 0, stream>>>(S, trc);
    init_kernel <<<(unsigned)(MAT / 256), 256, 0, stream>>>(S, trc, Ya, Za);

    float* T  = S;
    float* Y  = Ya; float* Y2 = Yb;
    float* Z  = Za; float* Z2 = Zb;
    for (int it = 0; it < 3; ++it) {
        // T = 3I - Z@Y
        gemm256_kernel<<<NBATCH * 4, 256, 0, stream>>>(Z, Y,  T,  -1.0f, 3.0f);
        // Y2 = 0.5 * Y@T
        gemm256_kernel<<<NBATCH * 4, 256, 0, stream>>>(Y, T,  Y2,  0.5f, 0.0f);
        // T = 3I - Z@Y2   (NOTE: reference uses the *updated* Y here)
        gemm256_kernel<<<NBATCH * 4, 256, 0, stream>>>(Z, Y2, T,  -1.0f, 3.0f);
        // Z2 = 0.5 * T@Z
        gemm256_kernel<<<NBATCH * 4, 256, 0, stream>>>(T, Z,  Z2,  0.5f, 0.0f);
        float* t0 = Y; Y = Y2; Y2 = t0;
        float* t1 = Z; Z = Z2; Z2 = t1;
    }

    final_kernel<<<dim3(CDIM, NBATCH), 256, 0, stream>>>(Y, trc, out);
}